// DefROIAlign_82669530514062
// MI455X (gfx1250) — compile-verified
//
#include <hip/hip_runtime.h>

// ---- problem constants (from reference) ----
#define DPH 7
#define DPW 7
static constexpr int   C_  = 256;
static constexpr int   H_  = 168;
static constexpr int   W_  = 100;
static constexpr float SCALE_     = 0.125f;
static constexpr float TRANS_STD_ = 0.1f;

// CDNA5 async-load-to-LDS path (guarded; falls back to plain LDS store)
#if defined(__gfx1250__) && __has_builtin(__builtin_amdgcn_global_load_async_to_lds_b32)
#define USE_ASYNC_LDS 1
#else
#define USE_ASYNC_LDS 0
#endif

#if defined(__gfx1250__) && __has_builtin(__builtin_amdgcn_s_wait_asynccnt)
#define WAIT_ASYNC() __builtin_amdgcn_s_wait_asynccnt(0)
#elif defined(__gfx1250__)
#define WAIT_ASYNC() asm volatile("s_wait_asynccnt 0" ::: "memory")
#else
#define WAIT_ASYNC()
#endif

__global__ __launch_bounds__(256) void def_roi_align_kernel(
    const float* __restrict__ x,        // (2,256,168,100) NCHW
    const float* __restrict__ rois,     // (R,5)
    const float* __restrict__ offsets,  // (R,2,7,7)
    float* __restrict__ out)            // (R,256,7,7)
{
  __shared__ float  sStage[8];  // [0..4]=roi row, [5]=off_w, [6]=off_h
  __shared__ int4   sOff[4];    // per-sample 4 corner offsets within a plane
  __shared__ float4 sW[4];      // per-sample 4 bilinear weights (pre-scaled by 1/4)

  const int bid = blockIdx.x;
  const int r   = bid / (DPH * DPW);
  const int pp  = bid % (DPH * DPW);
  const int ph  = pp / DPW;
  const int pw  = pp % DPW;
  const int tid = threadIdx.x;

  // ---- stage the 7 per-bin scalars into LDS (async DMA on CDNA5) ----
  if (tid < 7) {
    const float* g = (tid < 5)
        ? (rois + (size_t)r * 5 + tid)
        : (offsets + (((size_t)r * 2 + (tid - 5)) * DPH + ph) * DPW + pw);
#if USE_ASYNC_LDS
    typedef __attribute__((address_space(1))) int gint_t;
    typedef __attribute__((address_space(3))) int lint_t;
    __builtin_amdgcn_global_load_async_to_lds_b32(
        (gint_t*)g, (lint_t*)&sStage[tid], /*offset=*/0, /*cpol=*/0);
#else
    sStage[tid] = *g;
#endif
  }
#if USE_ASYNC_LDS
  WAIT_ASYNC();
#endif
  __syncthreads();

  // ---- threads 0..3: compute bilinear params for sample s = tid ----
  if (tid < 4) {
    const float rs_w  = sStage[1] * SCALE_ - 0.5f;
    const float rs_h  = sStage[2] * SCALE_ - 0.5f;
    const float roi_w = sStage[3] * SCALE_ - 0.5f - rs_w;
    const float roi_h = sStage[4] * SCALE_ - 0.5f - rs_h;
    const float bin_h = roi_h * (1.0f / DPH);
    const float bin_w = roi_w * (1.0f / DPW);
    const float d_w   = TRANS_STD_ * sStage[5] * roi_w;
    const float d_h   = TRANS_STD_ * sStage[6] * roi_h;

    const int   iy = tid >> 1, ix = tid & 1;
    const float gy = 0.25f + 0.5f * (float)iy;   // (iy+0.5)/SR, SR=2
    const float gx = 0.25f + 0.5f * (float)ix;

    const float ys = rs_h + d_h + ((float)ph + gy) * bin_h;
    const float xs = rs_w + d_w + ((float)pw + gx) * bin_w;

    const bool valid = (ys > -1.0f) && (ys < (float)H_) &&
                       (xs > -1.0f) && (xs < (float)W_);

    const float yc  = fmaxf(ys, 0.0f);
    const float xc  = fmaxf(xs, 0.0f);
    const int   yl0 = (int)floorf(yc);
    const int   xl0 = (int)floorf(xc);
    const int   yl  = min(yl0, H_ - 1);
    const int   xl  = min(xl0, W_ - 1);
    const int   yh  = min(yl0 + 1, H_ - 1);
    const int   xh  = min(xl0 + 1, W_ - 1);
    const float ly  = (yl0 >= H_ - 1) ? 0.0f : (yc - (float)yl0);
    const float lx  = (xl0 >= W_ - 1) ? 0.0f : (xc - (float)xl0);
    const float hy  = 1.0f - ly;
    const float hx  = 1.0f - lx;
    const float q   = valid ? 0.25f : 0.0f;     // fold mean over SR*SR=4

    sW[tid]   = make_float4(q * hy * hx, q * hy * lx, q * ly * hx, q * ly * lx);
    sOff[tid] = make_int4(yl * W_ + xl, yl * W_ + xh, yh * W_ + xl, yh * W_ + xh);
  }
  __syncthreads();

  // ---- per-channel gather + weighted accumulate ----
  const int    b     = (int)sStage[0];
  const int    c     = tid;
  const float* plane = x + ((size_t)(b * C_ + c)) * (size_t)(H_ * W_);

  // CDNA5 global_prefetch_b8: pull this channel's first target line toward L0/L2
  __builtin_prefetch(plane + sOff[0].x, 0, 0);

  float acc = 0.0f;
#pragma unroll
  for (int s = 0; s < 4; ++s) {
    const int4   o = sOff[s];
    const float4 w = sW[s];
    acc = fmaf(w.x, plane[o.x], acc);
    acc = fmaf(w.y, plane[o.y], acc);
    acc = fmaf(w.z, plane[o.z], acc);
    acc = fmaf(w.w, plane[o.w], acc);
  }

  out[(((size_t)r * C_ + c) * DPH + ph) * DPW + pw] = acc;
}

extern "C" void kernel_launch(void* const* d_in, const int* in_sizes, int n_in,
                              void* d_out, int out_size, void* d_ws, size_t ws_size,
                              hipStream_t stream) {
  (void)n_in; (void)out_size; (void)d_ws; (void)ws_size;
  const float* x       = (const float*)d_in[0];
  const float* rois    = (const float*)d_in[1];
  const float* offsets = (const float*)d_in[2];
  float*       out     = (float*)d_out;

  const int R = in_sizes[1] / 5;           // rois is (R,5)
  dim3 grid(R * DPH * DPW);                 // one block per output bin
  dim3 block(256);                          // one thread per channel (8 wave32)
  hipLaunchKernelGGL(def_roi_align_kernel, grid, block, 0, stream,
                     x, rois, offsets, out);
}